// MultiHeadLiftLayer_2937757631158
// MI455X (gfx1250) — compile-verified
//
#include <hip/hip_runtime.h>
#include <hip/hip_bf16.h>

typedef __attribute__((ext_vector_type(2))) float v2f;
typedef __attribute__((ext_vector_type(4))) float v4f;
typedef __attribute__((ext_vector_type(8))) float v8f;

// ---------------------------------------------------------------------------
// Kernel 1: fused projection  p[i][0..3] = x[i] . a_src[h],  p[i][4..7] = x[i] . a_tgt[h]
// One wave32 computes a 16x16 f32 tile via 16 chained V_WMMA_F32_16X16X4_F32
// (K = 64 in steps of 4). Columns 0..7 are the 8 valid outputs, 8..15 padded 0.
// ---------------------------------------------------------------------------
__global__ void mhl_proj_wmma(const float* __restrict__ x,
                              const float* __restrict__ att,
                              float* __restrict__ p,
                              int n_nodes) {
  const int wavesPerBlock = blockDim.x >> 5;
  const int wave = blockIdx.x * wavesPerBlock + (threadIdx.x >> 5);
  const int lane = threadIdx.x & 31;
  const int n_tiles = (n_nodes + 15) >> 4;
  if (wave >= n_tiles) return;                 // wave-uniform: EXEC stays all-1s

  const int row_base = wave << 4;
  const int m  = lane & 15;                    // M index (A) / N index (B,C,D)
  const int kh = lane >> 4;                    // selects K pair within a step

  // ---- A-matrix: row of x, clamped (tail rows recompute row n_nodes-1) ----
  int row = row_base + m;
  if (row >= n_nodes) row = n_nodes - 1;       // select, not branch
  const float* xrow = x + (long)row * 64;

  // ---- B-matrix: column n of W (16x64). W[0..3]=a_src, W[4..7]=a_tgt, rest 0.
  const int  n     = m;
  const bool valid = (n < 8);
  const float* wrow = valid ? ((n < 4) ? (att + n * 128)
                                       : (att + (n - 4) * 128 + 64))
                            : att;             // safe dummy address for padded cols

  v8f c = {};
  #pragma unroll
  for (int s = 0; s < 16; ++s) {
    const int off = 4 * s + 2 * kh;            // even -> 8B-aligned float2 loads
    v2f a = *(const v2f*)(xrow + off);
    v2f bl = *(const v2f*)(wrow + off);
    v2f b;
    b.x = valid ? bl.x : 0.0f;                 // zero-pad columns 8..15
    b.y = valid ? bl.y : 0.0f;
    // (neg_a, A, neg_b, B, c_mod, C, reuse_a, reuse_b)
    c = __builtin_amdgcn_wmma_f32_16x16x4_f32(false, a, false, b,
                                              (short)0, c, false, false);
  }

  // ---- C/D layout: lane l holds column n=l%16; VGPR v -> row v + 8*(l/16) ----
  if (valid) {
    #pragma unroll
    for (int v = 0; v < 8; ++v) {
      const int r = row_base + v + 8 * kh;
      if (r < n_nodes) p[(long)r * 8 + n] = c[v];
    }
  }
}

// ---------------------------------------------------------------------------
// Kernel 2: per-edge gather + relu.  p (1.6 MB) is L2-resident on MI455X, so
// the random gathers are L2 hits; HBM traffic is edge_index in + output out.
// ---------------------------------------------------------------------------
__global__ void mhl_edge_gather(const long long* __restrict__ src,
                                const long long* __restrict__ tgt,
                                const float* __restrict__ p,
                                float* __restrict__ out,
                                int n_edges) {
  const int e = blockIdx.x * blockDim.x + threadIdx.x;
  if (e >= n_edges) return;
  const long long s = src[e];
  const long long t = tgt[e];
  const v4f ps = *(const v4f*)(p + s * 8);      // cols 0..3  (src proj)
  const v4f pt = *(const v4f*)(p + t * 8 + 4);  // cols 4..7  (tgt proj)
  v4f o;
  o.x = fmaxf(ps.x + pt.x, 0.0f);
  o.y = fmaxf(ps.y + pt.y, 0.0f);
  o.z = fmaxf(ps.z + pt.z, 0.0f);
  o.w = fmaxf(ps.w + pt.w, 0.0f);
  __builtin_nontemporal_store(o, (v4f*)(out + (long)e * 4));
}

extern "C" void kernel_launch(void* const* d_in, const int* in_sizes, int n_in,
                              void* d_out, int out_size, void* d_ws, size_t ws_size,
                              hipStream_t stream) {
  const float*     x   = (const float*)d_in[0];        // (n_nodes, 64) f32
  const long long* ei  = (const long long*)d_in[1];    // (2, n_edges) i64
  const float*     att = (const float*)d_in[2];        // (4, 128) f32
  float* out = (float*)d_out;                          // (n_edges, 4) f32
  float* p   = (float*)d_ws;                           // (n_nodes, 8) f32 scratch

  const int n_nodes = in_sizes[0] / 64;
  const int n_edges = in_sizes[1] / 2;

  // Projection: 1 wave per 16 rows, 8 waves (256 threads) per block.
  const int n_tiles = (n_nodes + 15) / 16;
  const int blocks1 = (n_tiles + 7) / 8;
  mhl_proj_wmma<<<blocks1, 256, 0, stream>>>(x, att, p, n_nodes);

  // Edge gather: one thread per edge.
  const int threads = 256;
  const int blocks2 = (n_edges + threads - 1) / threads;
  mhl_edge_gather<<<blocks2, threads, 0, stream>>>(ei, ei + n_edges, p, out, n_edges);
}